// FastSpeech2TRT_45904610460090
// MI455X (gfx1250) — compile-verified
//
#include <hip/hip_runtime.h>
#include <cstdint>
#include <cstddef>

#define B_    16
#define S_    512
#define H_    384
#define L_    2000
#define NBINS 256

typedef uint32_t u32;
typedef __attribute__((ext_vector_type(4))) u32   u32x4;
typedef __attribute__((ext_vector_type(8))) int   i32x8;
typedef __attribute__((ext_vector_type(4))) int   i32x4;
typedef __attribute__((ext_vector_type(4))) float f32x4;

#if __has_include(<hip/amd_detail/amd_gfx1250_TDM.h>)
#define TDM_6ARG 1
#endif

__device__ inline void tdm_issue(u32x4 g0, i32x8 g1, i32x4 g2, i32x4 g3) {
#ifdef TDM_6ARG
    __builtin_amdgcn_tensor_load_to_lds(g0, g1, g2, g3, (i32x8)0, 0);
#else
    __builtin_amdgcn_tensor_load_to_lds(g0, g1, g2, g3, 0);
#endif
}

// Normal-mode TDM: contiguous 1-row tile of `nelem` 4-byte elements.
__device__ inline void tdm_load_1d(u32 lds_off, const void* gptr, u32 nelem) {
    uint64_t ga = (uint64_t)(uintptr_t)gptr;
    u32x4 g0;
    g0[0] = 1u;                                    // count=1, user mode, no gather
    g0[1] = lds_off;                               // lds_addr       [63:32]
    g0[2] = (u32)ga;                               // global_addr lo [95:64]
    g0[3] = ((u32)(ga >> 32) & 0x01FFFFFFu)        // global_addr hi [120:96]
          | (2u << 30);                            // type=2         [127:126]
    i32x8 g1;
    g1[0] = (int)(2u << 16);                       // data_size=2 (4B), wg_mask=0
    g1[1] = (int)((nelem & 0xFFFFu) << 16);        // tensor_dim0 lo16   [63:48]
    g1[2] = (int)(((nelem >> 16) & 0xFFFFu)        // tensor_dim0 hi16   [79:64]
          |  (1u << 16));                          // tensor_dim1=1 lo16 [111:96]... [111:80] lo
    g1[3] = (int)((nelem & 0xFFFFu) << 16);        // tile_dim0      [127:112]
    g1[4] = 1;                                     // tile_dim1=1    [143:128]
    g1[5] = (int)nelem;                            // tensor_dim0_stride lo32
    g1[6] = 0;  g1[7] = 0;
    tdm_issue(g0, g1, (i32x4)0, (i32x4)0);
}

// Gather-mode TDM: 16 rows of `width` 4-byte elements, 16-bit row indices in
// g2/g3, OOB rows (index >= nrows) are written as zeros by the TDM.
__device__ inline void tdm_gather16(u32 lds_off, const void* gbase, u32 width,
                                    u32 nrows, i32x4 g2, i32x4 g3) {
    uint64_t ga = (uint64_t)(uintptr_t)gbase;
    u32x4 g0;
    g0[0] = 1u | (1u << 31);                       // count=1, gather_mode=1, 16b idx
    g0[1] = lds_off;
    g0[2] = (u32)ga;
    g0[3] = ((u32)(ga >> 32) & 0x01FFFFFFu) | (2u << 30);
    i32x8 g1;
    g1[0] = (int)(2u << 16);                       // data_size=2 (4B)
    g1[1] = (int)((width & 0xFFFFu) << 16);        // tensor_dim0 lo16
    g1[2] = (int)(((width >> 16) & 0xFFFFu)        // tensor_dim0 hi16
          |  ((nrows & 0xFFFFu) << 16));           // tensor_dim1 lo16 (row OOB bound)
    g1[3] = (int)(((nrows >> 16) & 0xFFFFu)        // tensor_dim1 hi16
          |  ((width & 0xFFFFu) << 16));           // tile_dim0
    g1[4] = 16;                                    // tile_dim1 = #valid indices
    g1[5] = (int)width;                            // tensor_dim0_stride lo32
    g1[6] = 0;  g1[7] = 0;
    tdm_issue(g0, g1, g2, g3);
}

// ---------------------------------------------------------------------------
// Kernel 1: per-batch duration scan, mel_len, pitch/energy bin -> combined
// embedding row. One block per batch, 512 threads (16 waves).
// ---------------------------------------------------------------------------
__global__ __launch_bounds__(512) void fs2_prep_kernel(
    const float* __restrict__ logd, const float* __restrict__ pitch,
    const float* __restrict__ energy, const float* __restrict__ ptab,
    const float* __restrict__ etab, int* __restrict__ cum_ws,
    float* __restrict__ emb_ws, int* __restrict__ mel_out)
{
    __shared__ int   s_scan[S_];
    __shared__ float s_red[S_];
    __shared__ int   s_idx[2];

    const int b = blockIdx.x;
    const int t = threadIdx.x;

    // duration = max(rint(exp(log_d) - 1), 0)  (rint == round-half-even == jnp.round)
    float ld  = logd[b * S_ + t];
    int   dur = (int)fmaxf(rintf(expf(ld) - 1.0f), 0.0f);

    // Inclusive Hillis-Steele scan over 512 elements in LDS.
    s_scan[t] = dur;
    __syncthreads();
    for (int off = 1; off < S_; off <<= 1) {
        int v   = s_scan[t];
        int add = (t >= off) ? s_scan[t - off] : 0;
        __syncthreads();
        s_scan[t] = v + add;
        __syncthreads();
    }
    cum_ws[b * S_ + t] = s_scan[t];
    if (t == 0) {
        int total  = s_scan[S_ - 1];
        mel_out[b] = (total < L_) ? total : L_;
    }

    // Mean(pitch) -> clipped bin index
    s_red[t] = pitch[b * S_ + t];
    __syncthreads();
    for (int off = S_ / 2; off > 0; off >>= 1) {
        if (t < off) s_red[t] += s_red[t + off];
        __syncthreads();
    }
    if (t == 0) {
        int ip   = (int)(s_red[0] * (1.0f / (float)S_));
        s_idx[0] = ip < 0 ? 0 : (ip > NBINS - 1 ? NBINS - 1 : ip);
    }
    __syncthreads();

    // Mean(energy) -> clipped bin index
    s_red[t] = energy[b * S_ + t];
    __syncthreads();
    for (int off = S_ / 2; off > 0; off >>= 1) {
        if (t < off) s_red[t] += s_red[t + off];
        __syncthreads();
    }
    if (t == 0) {
        int ie   = (int)(s_red[0] * (1.0f / (float)S_));
        s_idx[1] = ie < 0 ? 0 : (ie > NBINS - 1 ? NBINS - 1 : ie);
    }
    __syncthreads();

    // Pre-combined per-batch embedding row (added to every output position).
    if (t < H_) {
        emb_ws[b * H_ + t] = ptab[s_idx[0] * H_ + t] + etab[s_idx[1] * H_ + t];
    }
}

// ---------------------------------------------------------------------------
// Kernel 2: length-regulator. Block = 16 mel positions x H=384, 256 threads.
// Wave 0: TDM-load cum+emb, binary-search the 16 source rows, pack indices
// into the D# and issue ONE gather-mode tensor_load_to_lds (16 rows, OOB rows
// -> zeros). All waves: out = tile + emb, streamed with non-temporal b128.
// ---------------------------------------------------------------------------
__global__ __launch_bounds__(256) void fs2_expand_kernel(
    const float* __restrict__ enc, const int* __restrict__ cum_ws,
    const float* __restrict__ emb_ws, float* __restrict__ out)
{
    __shared__ __align__(16) float s_tile[16 * H_];   // 24KB gathered rows
    __shared__ __align__(16) float s_emb[H_];
    __shared__ __align__(16) int   s_cum[S_];
    __shared__ int                 s_sidx[16];

    const int b   = blockIdx.y;
    const int l0  = blockIdx.x * 16;
    const int tid = threadIdx.x;

    if (tid < 32) {                                   // wave 0 only
        tdm_load_1d((u32)(uintptr_t)&s_cum[0], cum_ws + b * S_, S_);
        tdm_load_1d((u32)(uintptr_t)&s_emb[0], emb_ws + b * H_, H_);
        __builtin_amdgcn_s_wait_tensorcnt(0);

        // Lanes 0..15: smallest s with cum[s] > l (intervals partition [0,total)).
        int l = l0 + (tid & 15);
        int total = s_cum[S_ - 1];
        int s = -1;
        if (l < total) {
            int lo = 0, hi = S_;
            while (lo < hi) {
                int mid = (lo + hi) >> 1;
                if (s_cum[mid] > l) hi = mid; else lo = mid + 1;
            }
            s = lo;
        }
        if (tid < 16) s_sidx[tid] = s;                // (debug/visibility)

        // Row index for the TDM gather: OOB sentinel 0xFFFF => TDM writes zeros.
        // Indices are non-decreasing in l (gather-mode OOB requirement holds).
        u32 uv = (s < 0) ? 0xFFFFu : (u32)s;
        i32x4 g2, g3;
#pragma unroll
        for (int i = 0; i < 4; ++i) {
            u32 lo = (u32)__builtin_amdgcn_readlane((int)uv, 2 * i);
            u32 hi = (u32)__builtin_amdgcn_readlane((int)uv, 2 * i + 1);
            g2[i] = (int)(lo | (hi << 16));
        }
#pragma unroll
        for (int i = 0; i < 4; ++i) {
            u32 lo = (u32)__builtin_amdgcn_readlane((int)uv, 8 + 2 * i);
            u32 hi = (u32)__builtin_amdgcn_readlane((int)uv, 9 + 2 * i);
            g3[i] = (int)(lo | (hi << 16));
        }

        tdm_gather16((u32)(uintptr_t)&s_tile[0], enc + (size_t)b * S_ * H_,
                     H_, S_, g2, g3);
        __builtin_amdgcn_s_wait_tensorcnt(0);
    }
    __syncthreads();

    // 16 rows of one batch are contiguous in the output: one linear 24KB burst
    // per block. Output is write-once -> non-temporal stores keep L2 for enc.
    const f32x4* tile = (const f32x4*)s_tile;         // 1536 float4, row-major
    const f32x4* embv = (const f32x4*)s_emb;          // 96 float4
    f32x4* outv = (f32x4*)(out + ((size_t)b * L_ + l0) * H_);
#pragma unroll
    for (int k = 0; k < 6; ++k) {
        int v = tid + k * 256;                        // 0..1535
        int c = v - (v / 96) * 96;
        f32x4 r = tile[v] + embv[c];
        __builtin_nontemporal_store(r, outv + v);
    }
}

extern "C" void kernel_launch(void* const* d_in, const int* in_sizes, int n_in,
                              void* d_out, int out_size, void* d_ws, size_t ws_size,
                              hipStream_t stream) {
    const float* enc    = (const float*)d_in[0];   // (B,S,H)
    const float* logd   = (const float*)d_in[1];   // (B,S)
    const float* pitch  = (const float*)d_in[2];   // (B,S)
    const float* energy = (const float*)d_in[3];   // (B,S)
    const float* ptab   = (const float*)d_in[4];   // (NBINS,H)
    const float* etab   = (const float*)d_in[5];   // (NBINS,H)
    // d_in[6] = max_mel_len scalar (2000, compile-time constant here)

    float* out = (float*)d_out;                          // (B,L,H) f32
    int*   mel = (int*)(out + (size_t)B_ * L_ * H_);     // (B,) i32 tail

    int*   cum_ws = (int*)d_ws;                                      // 32KB
    float* emb_ws = (float*)((char*)d_ws + B_ * S_ * sizeof(int));   // 24KB

    fs2_prep_kernel<<<dim3(B_), dim3(S_), 0, stream>>>(
        logd, pitch, energy, ptab, etab, cum_ws, emb_ws, mel);

    fs2_expand_kernel<<<dim3(L_ / 16, B_), dim3(256), 0, stream>>>(
        enc, cum_ws, emb_ws, out);
}